// Attention_29515015258609
// MI455X (gfx1250) — compile-verified
//
#include <hip/hip_runtime.h>
#include <hip/hip_bf16.h>

#define BATCH       16
#define NUM_HEADS   32
#define NUM_KV      8
#define GQ          4            // NUM_HEADS / NUM_KV
#define HEAD_DIM    128
#define BLOCK_SIZE  256
#define MAX_BLOCKS  8
#define SQ_PITCH    132          // padded row pitch for Q in LDS (bank spread)
#define SP_PITCH    260          // padded row pitch for P in LDS (bank spread, 8B-aligned)
#define SV_PITCH    136          // V-tile pitch: 2*136 % 64 == 16 -> lane halves on disjoint banks
#define VTILE       32           // tokens per staged V tile
#define WS_PER      (2*GQ + GQ*HEAD_DIM)   // per (b,kv,chunk): m[4], l[4], acc[4][128] = 520 floats

typedef float v2f __attribute__((ext_vector_type(2)));
typedef float v8f __attribute__((ext_vector_type(8)));

__device__ __forceinline__ v8f wmma_f32(v2f a, v2f b, v8f c) {
  // V_WMMA_F32_16X16X4_F32: D = A(16x4) * B(4x16) + C(16x16), fp32 exact path
  return __builtin_amdgcn_wmma_f32_16x16x4_f32(false, a, false, b, (short)0, c, false, false);
}

// async DMA: global (64-bit vaddr) -> LDS (byte offset in VGPR), 16B, tracked by ASYNCcnt
__device__ __forceinline__ void async_copy_b128(unsigned lds_off, const void* gsrc) {
  asm volatile("global_load_async_to_lds_b128 %0, %1, off"
               :: "v"(lds_off), "v"((unsigned long long)(uintptr_t)gsrc)
               : "memory");
}
__device__ __forceinline__ void wait_async_le4() {
  asm volatile("s_wait_asynccnt 0x4" ::: "memory");
}
__device__ __forceinline__ void wait_async_0() {
  asm volatile("s_wait_asynccnt 0x0" ::: "memory");
}

// ---------------------------------------------------------------------------
// Kernel 1: per (chunk, kv, batch) flash-decoding partial.
//   grid = (MAX_BLOCKS, NUM_KV, BATCH), block = 256 (8 wave32)
// ---------------------------------------------------------------------------
__global__ __launch_bounds__(256)
void pa_partial(const float* __restrict__ q,  const float* __restrict__ kfreshp,
                const float* __restrict__ vfreshp, const float* __restrict__ kc,
                const float* __restrict__ vc, const int* __restrict__ bt,
                const int* __restrict__ ctx_lens, const int* __restrict__ slots,
                float* __restrict__ ws)
{
  const int chunk = blockIdx.x;
  const int kv    = blockIdx.y;
  const int b     = blockIdx.z;
  const int tid   = threadIdx.x;

  __shared__ float sQ[16][SQ_PITCH];             // heads 0..3 real, 4..15 zero
  __shared__ float sP[16][SP_PITCH];             // scores -> exp(P); heads x tokens
  __shared__ __align__(16) float sV[2][VTILE][SV_PITCH];  // double-buffered V tiles
  __shared__ float sRed[4][64];
  __shared__ float sM[4], sL[4];

  const int ctx = ctx_lens[b];
  float* wsp = ws + (size_t)((b * NUM_KV + kv) * MAX_BLOCKS + chunk) * WS_PER;
  const int chunk_start = chunk * BLOCK_SIZE;
  if (chunk_start >= ctx) {              // uniform across the block
    if (tid < GQ) { wsp[tid] = -1e30f; wsp[GQ + tid] = 0.0f; }
    return;
  }

  const int    blk        = bt[b * MAX_BLOCKS + chunk];
  const size_t row_stride = (size_t)NUM_KV * HEAD_DIM;   // 1024 floats between tokens
  const float* kc_base = kc + (size_t)blk * BLOCK_SIZE * row_stride + (size_t)kv * HEAD_DIM;
  const float* vc_base = vc + (size_t)blk * BLOCK_SIZE * row_stride + (size_t)kv * HEAD_DIM;
  const float* kf = kfreshp + ((size_t)b * NUM_KV + kv) * HEAD_DIM;
  const float* vf = vfreshp + ((size_t)b * NUM_KV + kv) * HEAD_DIM;
  // emulate the reference's scatter: if this chunk's block holds slot_mapping[b],
  // that token row comes from the fresh k/v instead of the cache
  const int sm = slots[b];
  const int fresh_tok = (sm >= 0 && (sm >> 8) == blk) ? (sm & (BLOCK_SIZE - 1)) : -1;

  // ---- stage Q (heads kv*4 .. kv*4+3) into LDS, zero-pad heads 4..15 ----
  for (int i = tid; i < 16 * SQ_PITCH; i += 256) ((float*)sQ)[i] = 0.0f;
  __syncthreads();
  {
    const float* qb = q + ((size_t)b * NUM_HEADS + kv * GQ) * HEAD_DIM;
    for (int e = tid; e < GQ * HEAD_DIM; e += 256)
      sQ[e / HEAD_DIM][e % HEAD_DIM] = qb[e];
  }
  __syncthreads();

  const int wave = tid >> 5;
  const int lane = tid & 31;
  const int lh   = lane & 15;     // N column / M row within half
  const int hi   = lane >> 4;     // lane half selects K pair {0,1} vs {2,3}
  const float scale = 0.08838834764831845f;   // 1/sqrt(128)

  // ============ Pass 1: S^T[tokens, heads] = K * Q^T via f32 WMMA ============
  for (int t = 0; t < 2; ++t) {
    const int tokA = wave * 32 + t * 16 + lh;     // A row = token
    const float* krow = (tokA == fresh_tok) ? kf
                       : (kc_base + (size_t)tokA * row_stride);
    v8f c = {0.f,0.f,0.f,0.f,0.f,0.f,0.f,0.f};
#pragma unroll 8
    for (int s = 0; s < 32; ++s) {
      v2f a  = *(const v2f*)(krow + 4 * s + 2 * hi);         // K dims 4s+{0,1}/{2,3}
      v2f bq = *(const v2f*)(&sQ[lh][4 * s + 2 * hi]);       // Q[head=lh][same dims]
      c = wmma_f32(a, bq, c);
    }
#pragma unroll
    for (int r = 0; r < 8; ++r) {
      const int tk = wave * 32 + t * 16 + 8 * hi + r;        // C row M = token
      sP[lh][tk] = (chunk_start + tk < ctx) ? c[r] * scale : -1e30f;
    }
  }

  // ---- kick off async stage of V tile 0 while scores/softmax proceed ----
  // tile = VTILE tokens x 128 floats; 512B/row = 32 x b128 segments; 4 per thread
  {
#pragma unroll
    for (int j = 0; j < 4; ++j) {
      const int seg  = tid + 256 * j;          // 0..1023
      const int row  = seg >> 5;               // 0..31 token in tile 0
      const int c16  = seg & 31;               // 16B segment within row
      const float* src = (row == fresh_tok) ? (vf + c16 * 4)
                        : (vc_base + (size_t)row * row_stride + c16 * 4);
      const unsigned lds = (unsigned)(uintptr_t)&sV[0][row][c16 * 4];
      async_copy_b128(lds, src);
    }
  }
  __syncthreads();   // sP complete for softmax

  // ============ chunk-local softmax: m, l, P = exp(s - m) ============
  const int h   = tid >> 6;      // 4 heads x 64 threads
  const int t64 = tid & 63;
  float mx = -1e30f;
#pragma unroll
  for (int i = 0; i < 4; ++i) mx = fmaxf(mx, sP[h][t64 + 64 * i]);
  sRed[h][t64] = mx;
  __syncthreads();
  for (int off = 32; off > 0; off >>= 1) {
    if (t64 < off) sRed[h][t64] = fmaxf(sRed[h][t64], sRed[h][t64 + off]);
    __syncthreads();
  }
  if (t64 == 0) sM[h] = sRed[h][0];
  __syncthreads();

  const float mh = sM[h];
  float psum = 0.0f;
#pragma unroll
  for (int i = 0; i < 4; ++i) {
    const int tk = t64 + 64 * i;
    const float e = __expf(sP[h][tk] - mh);
    sP[h][tk] = e;
    psum += e;
  }
  sRed[h][t64] = psum;
  __syncthreads();
  for (int off = 32; off > 0; off >>= 1) {
    if (t64 < off) sRed[h][t64] += sRed[h][t64 + off];
    __syncthreads();
  }
  if (t64 == 0) sL[h] = sRed[h][0];
  // zero P rows for the 12 padded heads so pass-2 A has clean zeros
  for (int r = GQ; r < 16; ++r) sP[r][tid] = 0.0f;
  __syncthreads();

  if (tid < GQ) { wsp[tid] = sM[tid]; wsp[GQ + tid] = sL[tid]; }

  // ============ Pass 2: acc[heads, 128] = P * V via f32 WMMA ============
  // V streamed through LDS in 8 double-buffered async tiles of 32 tokens.
  const int dcol = wave * 16 + lh;
  v8f acc = {0.f,0.f,0.f,0.f,0.f,0.f,0.f,0.f};

  for (int t = 0; t < BLOCK_SIZE / VTILE; ++t) {
    const int buf = t & 1;
    if (t + 1 < BLOCK_SIZE / VTILE) {
      const int nstart = (t + 1) * VTILE;
      const int nbuf   = (t + 1) & 1;
#pragma unroll
      for (int j = 0; j < 4; ++j) {
        const int seg  = tid + 256 * j;
        const int row  = seg >> 5;
        const int c16  = seg & 31;
        const int gtok = nstart + row;
        const float* src = (gtok == fresh_tok) ? (vf + c16 * 4)
                          : (vc_base + (size_t)gtok * row_stride + c16 * 4);
        const unsigned lds = (unsigned)(uintptr_t)&sV[nbuf][row][c16 * 4];
        async_copy_b128(lds, src);
      }
      wait_async_le4();       // tile t's 4 per-wave asyncs retired (in-order)
    } else {
      wait_async_0();
    }
    __syncthreads();          // whole tile visible to all waves

#pragma unroll
    for (int s = 0; s < VTILE / 4; ++s) {        // 4-token K-steps within tile
      const int lt0 = 4 * s + 2 * hi;            // local rows lt0, lt0+1
      const int t0  = t * VTILE + lt0;           // token index for P
      v2f a = *(const v2f*)(&sP[lh][t0]);        // P[head=lh][t0..t0+1]
      v2f bv; bv.x = sV[buf][lt0][dcol]; bv.y = sV[buf][lt0 + 1][dcol];
      acc = wmma_f32(a, bv, acc);
    }
    __syncthreads();          // done reading buf before it is refilled
  }

  if (hi == 0) {
#pragma unroll
    for (int r = 0; r < GQ; ++r)
      wsp[2 * GQ + r * HEAD_DIM + dcol] = acc[r];            // C row r = head r
  }
}

// ---------------------------------------------------------------------------
// Kernel 2: combine the 8 chunk partials per (b, kv) and normalize.
// ---------------------------------------------------------------------------
__global__ __launch_bounds__(256)
void pa_reduce(const float* __restrict__ ws, float* __restrict__ out)
{
  const int bkv = blockIdx.x;            // b*NUM_KV + kv
  const int b  = bkv / NUM_KV;
  const int kv = bkv % NUM_KV;
  const float* base = ws + (size_t)bkv * MAX_BLOCKS * WS_PER;

  for (int e = threadIdx.x; e < GQ * HEAD_DIM; e += 256) {
    const int h = e >> 7, d = e & (HEAD_DIM - 1);
    float M = -1e30f;
#pragma unroll
    for (int c = 0; c < MAX_BLOCKS; ++c) M = fmaxf(M, base[c * WS_PER + h]);
    float Ls = 0.0f, num = 0.0f;
#pragma unroll
    for (int c = 0; c < MAX_BLOCKS; ++c) {
      const float w = __expf(base[c * WS_PER + h] - M);      // 0 for invalid chunks
      Ls  += w * base[c * WS_PER + GQ + h];
      num += w * base[c * WS_PER + 2 * GQ + h * HEAD_DIM + d];
    }
    out[((size_t)b * NUM_HEADS + kv * GQ + h) * HEAD_DIM + d] = num / Ls;
  }
}

extern "C" void kernel_launch(void* const* d_in, const int* in_sizes, int n_in,
                              void* d_out, int out_size, void* d_ws, size_t ws_size,
                              hipStream_t stream) {
  (void)in_sizes; (void)n_in; (void)out_size; (void)ws_size;
  const float* q  = (const float*)d_in[0];
  const float* k  = (const float*)d_in[1];
  const float* v  = (const float*)d_in[2];
  const float* kc = (const float*)d_in[3];
  const float* vc = (const float*)d_in[4];
  const int*   bt = (const int*)d_in[5];
  const int*   cl = (const int*)d_in[6];
  const int*   sm = (const int*)d_in[7];
  float* out = (float*)d_out;
  float* ws  = (float*)d_ws;

  dim3 grid(MAX_BLOCKS, NUM_KV, BATCH);          // 1024 workgroups
  pa_partial<<<grid, 256, 0, stream>>>(q, k, v, kc, vc, bt, cl, sm, ws);
  pa_reduce<<<BATCH * NUM_KV, 256, 0, stream>>>(ws, out);
}